// ButterflyNewMlp_47304769798188
// MI455X (gfx1250) — compile-verified
//
#include <hip/hip_runtime.h>
#include <math.h>

// ---------------------------------------------------------------------------
// ButterflyNewMlp on MI455X (gfx1250, wave32).
//   attn = (w1^T @ w2^T) * butterfly_mask   -> only ~4% of 16x16 tiles nonzero
//   out  = gelu_exact(X @ attn + b2)
// Tile-sparse f32 WMMA (v_wmma_f32_16x16x4_f32). After sparsity the problem is
// HBM-bandwidth bound (~330MB @ 23.3TB/s) so full f32 precision is free.
// w2 staging uses CDNA5 async global->LDS (double buffered, ASYNCcnt); attn
// tiles are stored pre-transposed so kernel B's staging is a straight row copy.
// Workspace: flags[65536] (int) + attnT[4096*4096] (f32, [t][s] layout).
// ---------------------------------------------------------------------------

typedef float v2f __attribute__((ext_vector_type(2)));
typedef float v8f __attribute__((ext_vector_type(8)));
typedef int   v4i __attribute__((__vector_size__(4 * sizeof(int))));

#define NF      4096      // in = out = hid
#define MROWS   6144      // 8 * 768
#define NT      256       // 16x16 tiles per matrix dim

#if defined(__gfx1250__) && \
    __has_builtin(__builtin_amdgcn_global_load_async_to_lds_b128) && \
    __has_builtin(__builtin_amdgcn_s_wait_asynccnt)
#define USE_ASYNC_LDS 1
typedef __attribute__((address_space(1))) v4i* gptr4;
typedef __attribute__((address_space(3))) v4i* lptr4;
#else
#define USE_ASYNC_LDS 0
#endif

__device__ __forceinline__ void async_copy16(const float* src, float* lds_dst) {
#if USE_ASYNC_LDS
    __builtin_amdgcn_global_load_async_to_lds_b128(
        (gptr4)src, (lptr4)lds_dst, 0, 0);
#else
    *(float4*)lds_dst = *(const float4*)src;
#endif
}

__device__ __forceinline__ void wait_async_lds() {
#if USE_ASYNC_LDS
    __builtin_amdgcn_s_wait_asynccnt(0);
#endif
}

__device__ __forceinline__ float gelu_exact(float v) {
    return 0.5f * v * (1.0f + erff(v * 0.70710678118654752f));
}

// Stage w2[t0..t0+15][dbase..dbase+255] (16KB) into a padded LDS buffer.
// 128 threads x 8 x b128; async path issues 8 ASYNC ops per thread.
__device__ __forceinline__ void stage16x256(const float* __restrict__ src_row0,
                                            float* __restrict__ dst, int tid) {
    const int r  = tid >> 3;            // 0..15 (t-local)
    const int cb = (tid & 7) * 32;      // 0..224 (d-local)
    const float* src = src_row0 + (size_t)r * NF + cb;
    float* d = dst + r * 260 + cb;
    #pragma unroll
    for (int j = 0; j < 8; ++j)
        async_copy16(src + j * 4, d + j * 4);
}

// ---------------------------------------------------------------------------
// Kernel A: one workgroup (4 waves) per 16x16 attn tile.
//   - mask-tile vote -> early exit for ~96% of tiles (writes flag=0)
//   - split-K over 4 waves, K=4096, WMMA f32 16x16x4
//   - A (w1) loads coalesced from global (lane = s column)
//   - B (w2) double-buffered in LDS via async global->LDS
//   - result stored TRANSPOSED: attnT[t][s] (scattered, but tiny volume)
// ---------------------------------------------------------------------------
__global__ __launch_bounds__(128)
void attn_tiles_kernel(const float* __restrict__ w1,
                       const float* __restrict__ w2,
                       const float* __restrict__ mask,
                       float* __restrict__ attnT,
                       int* __restrict__ flags) {
    __shared__ float lds_w2[2][16 * 260];   // double-buffered, padded
    __shared__ float cred[4 * 8 * 32];      // split-K partials
    __shared__ int   lds_nz;

    const int tile = blockIdx.x;
    const int si = tile >> 8, ti = tile & 255;
    const int s0 = si * 16, t0 = ti * 16;
    const int tid  = threadIdx.x;
    const int lane = tid & 31, wv = tid >> 5;
    const int lhi  = lane >> 4, llo = lane & 15;

    if (tid == 0) lds_nz = 0;
    __syncthreads();
    {   // 128 threads x 2 floats cover the 16x16 mask tile (coalesced)
        const int r = tid >> 3, c2 = (tid & 7) * 2;
        v2f mv = *(const v2f*)(mask + (size_t)(s0 + r) * NF + t0 + c2);
        if (mv[0] != 0.0f || mv[1] != 0.0f) atomicOr(&lds_nz, 1);
    }
    __syncthreads();
    const int nz = lds_nz;
    if (tid == 0) flags[tile] = nz;
    if (!nz) return;

    v8f c = {0.f, 0.f, 0.f, 0.f, 0.f, 0.f, 0.f, 0.f};

    const float* w2row = w2 + (size_t)t0 * NF;
    stage16x256(w2row, &lds_w2[0][0], tid);              // prefetch chunk 0

    for (int ch = 0; ch < 16; ++ch) {                    // K chunks of 256
        wait_async_lds();                                // my async writes done
        __syncthreads();                                 // everyone's done
        if (ch + 1 < 16)                                 // prefetch next chunk
            stage16x256(w2row + (ch + 1) * 256, &lds_w2[(ch + 1) & 1][0], tid);

        const float* lw = &lds_w2[ch & 1][0];
        const int dbase = ch * 256;
        const int kw = wv * 64;                          // split-K: 64 d / wave
        #pragma unroll
        for (int kk = 0; kk < 16; ++kk) {
            const int klocal = kw + kk * 4;
            const int kv = klocal + 2 * lhi;             // K halves {0,1}/{2,3}
            // A[m][k] = w1[d][s0+m]: lane=column -> coalesced b32 pair
            const size_t a0i = (size_t)(dbase + kv) * NF + s0 + llo;
            v2f a; a[0] = w1[a0i]; a[1] = w1[a0i + NF];
            // B[k][n] = w2[t0+n][d]: one ds_load_b64 feeds both B VGPRs
            v2f b = *(const v2f*)(lw + llo * 260 + kv);
            c = __builtin_amdgcn_wmma_f32_16x16x4_f32(
                    false, a, false, b, (short)0, c, false, false);
        }
        // no trailing barrier: next iteration's wait+barrier fences the swap
    }

    // split-K reduction through LDS
    #pragma unroll
    for (int r = 0; r < 8; ++r) cred[(wv * 8 + r) * 32 + lane] = c[r];
    __syncthreads();
    if (wv == 0) {
        #pragma unroll
        for (int r = 0; r < 8; ++r) {
            float s = cred[r * 32 + lane] + cred[(8 + r) * 32 + lane]
                    + cred[(16 + r) * 32 + lane] + cred[(24 + r) * 32 + lane];
            // mask read coalesced (D layout: M=r+8*lhi, N=llo)
            const int srow = s0 + r + 8 * lhi;
            const size_t om = (size_t)srow * NF + t0 + llo;
            // store transposed: attnT[t][s]
            attnT[(size_t)(t0 + llo) * NF + srow] = s * mask[om];
        }
    }
}

// ---------------------------------------------------------------------------
// Kernel B: workgroup = 4 waves = 64 M-rows x 16 t-cols of the output.
// Skips zero s-tiles via flags (~8 of 256 survive). attnT tile staged via a
// straight async row copy (no transpose needed), ds_load_b64 feeds both B
// VGPRs per WMMA. Epilogue: bias + exact-erf GELU, coalesced stores.
// ---------------------------------------------------------------------------
__global__ __launch_bounds__(128)
void mlp_out_kernel(const float* __restrict__ x,
                    const float* __restrict__ attnT,
                    const int* __restrict__ flags,
                    const float* __restrict__ b2,
                    float* __restrict__ out) {
    __shared__ float att_t[16 * 20];   // [t-local][s-local], padded stride 20

    const int tid  = threadIdx.x;
    const int lane = tid & 31, wv = tid >> 5;
    const int lhi  = lane >> 4, llo = lane & 15;
    const int ti = blockIdx.y;
    const int t0 = ti * 16;
    const int m0 = (blockIdx.x * 4 + wv) * 16;

    v8f c = {0.f, 0.f, 0.f, 0.f, 0.f, 0.f, 0.f, 0.f};

    for (int si = 0; si < NT; ++si) {
        if (!flags[si * NT + ti]) continue;      // WG-uniform -> safe barriers
        const int s0 = si * 16;
        __syncthreads();                          // prior-iter reads done
        if (tid < 64) {                           // 64 threads x b128 = 1KB tile
            const int r  = tid >> 2;              // t-local
            const int sg = (tid & 3) * 4;         // s-local segment
            async_copy16(attnT + (size_t)(t0 + r) * NF + s0 + sg,
                         att_t + r * 20 + sg);
        }
        wait_async_lds();
        __syncthreads();
        #pragma unroll
        for (int kk = 0; kk < 4; ++kk) {
            const int kb = kk * 4 + 2 * lhi;
            // A[m][k] = x[m0+m][s0+k]: b64 per lane (consecutive k)
            v2f a = *(const v2f*)(x + (size_t)(m0 + llo) * NF + s0 + kb);
            // B[k][n] = attnT[t0+n][s0+k] = att_t[n][k]
            v2f b = *(const v2f*)(att_t + llo * 20 + kb);
            c = __builtin_amdgcn_wmma_f32_16x16x4_f32(
                    false, a, false, b, (short)0, c, false, false);
        }
    }

    const float bias = b2[t0 + llo];
    #pragma unroll
    for (int r = 0; r < 8; ++r) {
        const float v = c[r] + bias;
        out[(size_t)(m0 + r + 8 * lhi) * NF + t0 + llo] = gelu_exact(v);
    }
}

// ---------------------------------------------------------------------------
extern "C" void kernel_launch(void* const* d_in, const int* in_sizes, int n_in,
                              void* d_out, int out_size, void* d_ws, size_t ws_size,
                              hipStream_t stream) {
    const float* x    = (const float*)d_in[0];   // [8,768,4096] -> [6144,4096]
    const float* w1   = (const float*)d_in[1];   // [4096,4096] (hid,in)
    const float* w2   = (const float*)d_in[2];   // [4096,4096] (out,hid)
    const float* b2   = (const float*)d_in[3];   // [4096]
    const float* mask = (const float*)d_in[4];   // [4096,4096]
    float* out = (float*)d_out;

    int*   flags = (int*)d_ws;                             // 65536 ints
    float* attnT = (float*)((char*)d_ws + (size_t)NT * NT * sizeof(int));

    attn_tiles_kernel<<<NT * NT, 128, 0, stream>>>(w1, w2, mask, attnT, flags);

    dim3 gridB(MROWS / 64, NT);
    mlp_out_kernel<<<gridB, 128, 0, stream>>>(x, attnT, flags, b2, out);
}